// YoloNASPoseTaskAlignedAssigner_11364483465643
// MI455X (gfx1250) — compile-verified
//
#include <hip/hip_runtime.h>
#include <cstddef>

#define TOPK 13
#define EPSV 1e-9f
#define MAXK 32   // K=17 at runtime; LDS stride padded to 32

typedef float v2f __attribute__((ext_vector_type(2)));
typedef float v8f __attribute__((ext_vector_type(8)));

// ---------------------------------------------------------------------------
// Kernel 1: init scratch (topk count bytes, per-instance max accumulators)
// ---------------------------------------------------------------------------
__global__ void k_init(unsigned char* __restrict__ intk,
                       unsigned int* __restrict__ maxmet,
                       unsigned int* __restrict__ maxiou,
                       size_t nbytes, int BN)
{
    size_t i = (size_t)blockIdx.x * blockDim.x + threadIdx.x;
    size_t stride = (size_t)gridDim.x * blockDim.x;
    for (size_t p = i; p < nbytes; p += stride) intk[p] = 0;
    if (i < (size_t)BN) { maxmet[i] = 0u; maxiou[i] = 0u; }
}

// ---------------------------------------------------------------------------
// Kernel 2: per 16x16 (gt x anchor) tile — OKS via V_WMMA_F32_16X16X4_F32
// cross terms, box IoU, alignment metric, in-gts mask.
// One wave (32 threads) per tile; EXEC all ones as WMMA requires.
// Inner loop: 1 WMMA + fused weight multiply + v_exp per element; the
// per-keypoint 1/(8*sigma^2) divide and the 1/(area+eps) factor are
// pre-combined into LDS weights s_w[row][k] during the tile-load phase.
// ---------------------------------------------------------------------------
__global__ __launch_bounds__(32)
void k_metrics(const float* __restrict__ pred_scores,
               const float* __restrict__ pred_bboxes,
               const float* __restrict__ pred_kpts,
               const float* __restrict__ anchors,
               const int*   __restrict__ gt_labels,
               const float* __restrict__ gt_bboxes,
               const float* __restrict__ gt_poses,
               const float* __restrict__ sigmas,
               float* __restrict__ ious_ws,
               float* __restrict__ met_ws,
               unsigned char* __restrict__ ingts_ws,
               int B, int L, int C, int N, int K)
{
    __shared__ float s_kx[16*MAXK], s_ky[16*MAXK], s_gsq[16*MAXK], s_vis[16*MAXK];
    __shared__ float s_w[16*MAXK];           // invsig[k] * invarea[row]
    __shared__ float s_invsig[MAXK];
    __shared__ float s_gb[16*4];
    __shared__ float s_invarea[16], s_nvinv[16];
    __shared__ int   s_lab[16];

    const int b   = blockIdx.z;
    const int n0  = blockIdx.y * 16;
    const int l0  = blockIdx.x * 16;
    const int tid = threadIdx.x;

    // cooperative load of 16 GT rows into LDS
    for (int i = tid; i < 16 * K; i += 32) {
        int r = i / K, kk = i % K;
        int n = n0 + r;
        float kx = 0.f, ky = 0.f, vis = 0.f;
        if (n < N) {
            const float* gp = gt_poses + ((size_t)(b * N + n) * K + kk) * 3;
            kx = gp[0]; ky = gp[1]; vis = (gp[2] > 0.f) ? 1.f : 0.f;
        }
        s_kx[r*MAXK+kk] = kx; s_ky[r*MAXK+kk] = ky;
        s_gsq[r*MAXK+kk] = kx*kx + ky*ky; s_vis[r*MAXK+kk] = vis;
    }
    if (tid < K) {                              // per-keypoint 1/(8 s^2), once
        float sg = sigmas[tid];
        s_invsig[tid] = 1.f / (8.f * sg * sg);
    }
    if (tid < 16) {
        int n = n0 + tid;
        float x1=0,y1=0,x2=0,y2=0; int lab = 0;
        if (n < N) {
            const float* gb = gt_bboxes + (size_t)(b * N + n) * 4;
            x1 = gb[0]; y1 = gb[1]; x2 = gb[2]; y2 = gb[3];
            lab = gt_labels[b * N + n];
            if (lab < 0) lab = 0;
            if (lab > C - 1) lab = C - 1;
        }
        s_gb[tid*4+0]=x1; s_gb[tid*4+1]=y1; s_gb[tid*4+2]=x2; s_gb[tid*4+3]=y2;
        float area = (x2 - x1) * (y2 - y1) * 0.53f;
        s_invarea[tid] = (n < N) ? 1.f / (area + EPSV) : 0.f;
        s_lab[tid] = lab;
    }
    __syncthreads();
    if (tid < 16) {
        float nv = 0.f;
        float ia = s_invarea[tid];
        for (int kk = 0; kk < K; ++kk) {
            nv += s_vis[tid*MAXK+kk];
            s_w[tid*MAXK+kk] = s_invsig[kk] * ia;   // combined OKS weight
        }
        s_nvinv[tid] = 1.f / (nv + EPSV);
    }
    __syncthreads();

    const int c    = tid & 15;
    const int half = tid >> 4;
    int  l      = l0 + c;
    bool lvalid = (l < L);
    int  lc     = lvalid ? l : (L - 1);

    float ax = anchors[(size_t)lc*2+0], ay = anchors[(size_t)lc*2+1];
    const float* pb = pred_bboxes + (size_t)(b * L + lc) * 4;
    float px1 = pb[0], py1 = pb[1], px2 = pb[2], py2 = pb[3];
    float wp = px2 - px1; if (wp < 0.f) wp = 0.f;
    float hp = py2 - py1; if (hp < 0.f) hp = 0.f;
    float area_p = wp * hp;

    float okssum[8];
#pragma unroll
    for (int j = 0; j < 8; ++j) okssum[j] = 0.f;

    const float* pk_base = pred_kpts + (size_t)(b * L + lc) * K * 2;

    for (int kk = 0; kk < K; ++kk) {
        const float* pk = pk_base + (size_t)kk * 2;
        if (kk + 1 < K)
            __builtin_prefetch(pk_base + (size_t)(kk + 1) * 2, 0, 0);
        float px = pk[0], py = pk[1];
        float psq = px*px + py*py;

        // A (16x4 f32): lanes 0-15 hold K-slots 0/1 = (gx,gy) of row m=lane;
        // lanes 16-31 hold K-slots 2/3 = 0.  B (4x16 f32): lanes 0-15 hold
        // K-rows 0/1 = (px,py) of col c=lane; lanes 16-31 (K-rows 2/3) = 0.
        v2f av, bv;
        if (half == 0) {
            av.x = s_kx[c*MAXK+kk]; av.y = s_ky[c*MAXK+kk];
            bv.x = px;              bv.y = py;
        } else {
            av.x = 0.f; av.y = 0.f; bv.x = 0.f; bv.y = 0.f;
        }
        v8f czero = {};
        v8f cross = __builtin_amdgcn_wmma_f32_16x16x4_f32(
            false, av, false, bv, (short)0, czero, false, false);

#pragma unroll
        for (int j = 0; j < 8; ++j) {
            int row = j + 8 * half;             // D layout: VGPR j -> rows j / j+8
            float d = s_gsq[row*MAXK+kk] + psq - 2.f * cross[j];
            float e = d * s_w[row*MAXK+kk];
            okssum[j] += __expf(-e) * s_vis[row*MAXK+kk];
        }
    }

#pragma unroll
    for (int j = 0; j < 8; ++j) {
        int row = j + 8 * half;
        int n = n0 + row;
        float oks = okssum[j] * s_nvinv[row];

        float gx1 = s_gb[row*4+0], gy1 = s_gb[row*4+1];
        float gx2 = s_gb[row*4+2], gy2 = s_gb[row*4+3];
        float ox = fminf(gx2, px2) - fmaxf(gx1, px1); if (ox < 0.f) ox = 0.f;
        float oy = fminf(gy2, py2) - fmaxf(gy1, py1); if (oy < 0.f) oy = 0.f;
        float ov = ox * oy;
        float wg = gx2 - gx1; if (wg < 0.f) wg = 0.f;
        float hg = gy2 - gy1; if (hg < 0.f) hg = 0.f;
        float area_g = wg * hg;
        float iou = ov / (area_g + area_p - ov + EPSV);
        iou *= oks;                             // combined quality

        float sc = pred_scores[(size_t)(b * L + lc) * C + s_lab[row]];
        float iou2 = iou * iou;
        float metric = sc * (iou2 * iou2 * iou2);   // ALPHA=1, BETA=6

        float dmin = fminf(fminf(ax - gx1, ay - gy1), fminf(gx2 - ax, gy2 - ay));
        unsigned char flag = (dmin > EPSV) ? 1 : 0;

        if (n < N && lvalid) {
            size_t idx = ((size_t)(b * N + n)) * L + l;
            ious_ws[idx] = iou;
            met_ws[idx]  = metric;
            ingts_ws[idx] = flag;
        }
    }
}

// ---------------------------------------------------------------------------
// Kernel 3: top-13 over L per (b,n) row of metric*in_gts
// ---------------------------------------------------------------------------
__global__ __launch_bounds__(128)
void k_topk(const float* __restrict__ met,
            const unsigned char* __restrict__ ingts,
            const float* __restrict__ pad,
            int* __restrict__ topk, int L)
{
    int row = blockIdx.x;          // b*N + n
    int t   = threadIdx.x;
    float v[TOPK]; int id[TOPK];
#pragma unroll
    for (int s = 0; s < TOPK; ++s) { v[s] = -1.f; id[s] = 0; }

    size_t base = (size_t)row * L;
    for (int l = t; l < L; l += 128) {
        float x = ingts[base + l] ? met[base + l] : 0.f;
        if (x > v[TOPK-1]) {
            int pos = TOPK - 1;
            while (pos > 0 && v[pos-1] < x) {
                v[pos] = v[pos-1]; id[pos] = id[pos-1]; --pos;
            }
            v[pos] = x; id[pos] = l;
        }
    }
    __shared__ float sv[128 * TOPK];
    __shared__ int   si[128 * TOPK];
#pragma unroll
    for (int s = 0; s < TOPK; ++s) { sv[t*TOPK+s] = v[s]; si[t*TOPK+s] = id[s]; }
    __syncthreads();

    if (t == 0) {
        float padv = pad[row];
        for (int s = 0; s < TOPK; ++s) {
            float best = -2.f; int bi = 0;
            for (int i = 0; i < 128 * TOPK; ++i)
                if (sv[i] > best) { best = sv[i]; bi = i; }
            int outl = si[bi];
            sv[bi] = -3.f;
            topk[row * TOPK + s] = (padv > 0.f) ? outl : 0;
        }
    }
}

// ---------------------------------------------------------------------------
// Kernel 4: scatter top-k indices into count array (count>1 later treated 0)
// ---------------------------------------------------------------------------
__global__ void k_scatter(const int* __restrict__ topk,
                          unsigned char* __restrict__ intk,
                          int BN, int L)
{
    int row = blockIdx.x * blockDim.x + threadIdx.x;
    if (row >= BN) return;
    size_t base = (size_t)row * L;
    for (int s = 0; s < TOPK; ++s) {
        int l = topk[row * TOPK + s];
        if (l >= 0 && l < L) intk[base + l] = (unsigned char)(intk[base + l] + 1);
    }
}

// ---------------------------------------------------------------------------
// Kernel 5: per (b,l) — resolve mask_positive, pick assigned GT, feed
// per-instance max(metric)/max(iou) via non-negative-float atomicMax.
// ---------------------------------------------------------------------------
__global__ void k_assign(const float* __restrict__ ious,
                         const float* __restrict__ met,
                         const unsigned char* __restrict__ ingts,
                         const unsigned char* __restrict__ intk,
                         const float* __restrict__ pad,
                         int* __restrict__ nstar,
                         unsigned int* __restrict__ maxmet,
                         unsigned int* __restrict__ maxiou,
                         int B, int L, int N)
{
    int gid = blockIdx.x * blockDim.x + threadIdx.x;
    if (gid >= B * L) return;
    int b = gid / L, l = gid % L;

    int cnt = 0, first = -1, bestn = 0;
    float bestiou = -1.f;
    for (int n = 0; n < N; ++n) {
        size_t idx = ((size_t)(b * N + n)) * L + l;
        float iv = ious[idx];
        if (iv > bestiou) { bestiou = iv; bestn = n; }   // first-max (argmax)
        if (intk[idx] == 1 && ingts[idx] && pad[b * N + n] > 0.f) {
            ++cnt; if (first < 0) first = n;
        }
    }
    int ns = (cnt > 1) ? bestn : ((cnt == 1) ? first : -1);
    nstar[gid] = ns;
    if (ns >= 0) {
        size_t idx = ((size_t)(b * N + ns)) * L + l;
        atomicMax(&maxmet[b * N + ns], __float_as_uint(met[idx]));
        atomicMax(&maxiou[b * N + ns], __float_as_uint(ious[idx]));
    }
}

// ---------------------------------------------------------------------------
// Kernel 6: gather/finalize outputs (tuple concatenated flat as float)
// ---------------------------------------------------------------------------
__global__ void k_final(const int* __restrict__ nstar,
                        const float* __restrict__ met,
                        const unsigned int* __restrict__ maxmet,
                        const unsigned int* __restrict__ maxiou,
                        const int*   __restrict__ gt_labels,
                        const float* __restrict__ gt_bboxes,
                        const float* __restrict__ gt_poses,
                        const int*   __restrict__ gt_crowd,
                        const int*   __restrict__ bgp,
                        float* __restrict__ out,
                        int B, int L, int C, int N, int K)
{
    int gid = blockIdx.x * blockDim.x + threadIdx.x;
    if (gid >= B * L) return;
    int b = gid / L, l = gid % L;
    int bg = *bgp;

    int ns = nstar[gid];
    int g  = (ns >= 0) ? ns : 0;            // argmax of all-zero mask -> 0
    int crowd = gt_crowd[b * N + g];
    int lab = (ns >= 0) ? gt_labels[b * N + g] : bg;
    if (crowd > 0) lab = bg;

    float s = 0.f;
    if (ns >= 0) {
        size_t idx = ((size_t)(b * N + ns)) * L + l;
        float mm = __uint_as_float(maxmet[b * N + ns]);
        float mi = __uint_as_float(maxiou[b * N + ns]);
        s = met[idx] / (mm + EPSV) * mi;
    }
    if (crowd > 0) s = 0.f;

    size_t BL = (size_t)B * L;
    float* o_lab  = out;
    float* o_bb   = o_lab  + BL;
    float* o_pose = o_bb   + BL * 4;
    float* o_sc   = o_pose + BL * (size_t)K * 3;
    float* o_gi   = o_sc   + BL * C;
    float* o_cr   = o_gi   + BL;

    o_lab[gid] = (float)lab;
    const float* gb = gt_bboxes + (size_t)(b * N + g) * 4;
#pragma unroll
    for (int i = 0; i < 4; ++i) o_bb[(size_t)gid * 4 + i] = gb[i];
    const float* gp = gt_poses + (size_t)(b * N + g) * K * 3;
    for (int i = 0; i < K * 3; ++i) o_pose[(size_t)gid * K * 3 + i] = gp[i];
    for (int j = 0; j < C; ++j) {
        int cls = j + ((j >= bg) ? 1 : 0);  // kept class ids skip bg_index
        o_sc[(size_t)gid * C + j] = (lab == cls) ? s : 0.f;
    }
    o_gi[gid] = (float)(g + b * N);
    o_cr[gid] = (crowd > 0) ? 1.f : 0.f;
}

// ---------------------------------------------------------------------------
extern "C" void kernel_launch(void* const* d_in, const int* in_sizes, int n_in,
                              void* d_out, int out_size, void* d_ws, size_t ws_size,
                              hipStream_t stream)
{
    const float* pred_scores = (const float*)d_in[0];
    const float* pred_bboxes = (const float*)d_in[1];
    const float* pred_kpts   = (const float*)d_in[2];
    const float* anchors     = (const float*)d_in[3];
    const int*   gt_labels   = (const int*)d_in[4];
    const float* gt_bboxes   = (const float*)d_in[5];
    const float* gt_poses    = (const float*)d_in[6];
    const int*   gt_crowd    = (const int*)d_in[7];
    const float* pad_mask    = (const float*)d_in[8];
    const float* sigmas      = (const float*)d_in[9];
    const int*   bgp         = (const int*)d_in[10];

    const int K = in_sizes[9];
    const int L = in_sizes[3] / 2;
    const int B = in_sizes[1] / (L * 4);
    const int C = in_sizes[0] / (B * L);
    const int N = in_sizes[8] / B;
    const int BN = B * N;
    const size_t BNL = (size_t)B * N * L;

    auto a256 = [](size_t x) { return (x + 255) & ~(size_t)255; };
    char* w = (char*)d_ws;
    float*         ious_ws = (float*)w;         w += a256(BNL * 4);
    float*         met_ws  = (float*)w;         w += a256(BNL * 4);
    unsigned char* ingts   = (unsigned char*)w; w += a256(BNL);
    unsigned char* intk    = (unsigned char*)w; w += a256(BNL);
    int*           topk    = (int*)w;           w += a256((size_t)BN * TOPK * 4);
    int*           nstar   = (int*)w;           w += a256((size_t)B * L * 4);
    unsigned int*  maxmet  = (unsigned int*)w;  w += a256((size_t)BN * 4);
    unsigned int*  maxiou  = (unsigned int*)w;  w += a256((size_t)BN * 4);

    k_init<<<2048, 256, 0, stream>>>(intk, maxmet, maxiou, BNL, BN);

    dim3 mg((L + 15) / 16, (N + 15) / 16, B);
    k_metrics<<<mg, 32, 0, stream>>>(pred_scores, pred_bboxes, pred_kpts, anchors,
                                     gt_labels, gt_bboxes, gt_poses, sigmas,
                                     ious_ws, met_ws, ingts, B, L, C, N, K);

    k_topk<<<BN, 128, 0, stream>>>(met_ws, ingts, pad_mask, topk, L);

    k_scatter<<<(BN + 63) / 64, 64, 0, stream>>>(topk, intk, BN, L);

    k_assign<<<(B * L + 255) / 256, 256, 0, stream>>>(ious_ws, met_ws, ingts, intk,
                                                      pad_mask, nstar, maxmet, maxiou,
                                                      B, L, N);

    k_final<<<(B * L + 255) / 256, 256, 0, stream>>>(nstar, met_ws, maxmet, maxiou,
                                                     gt_labels, gt_bboxes, gt_poses,
                                                     gt_crowd, bgp, (float*)d_out,
                                                     B, L, C, N, K);
}